// DSDM_2851858284940
// MI455X (gfx1250) — compile-verified
//
#include <hip/hip_runtime.h>
#include <hip/hip_bf16.h>
#include <math.h>

// DSDM soft retrieval for MI455X (gfx1250, wave32).
//   addresses: [N=131072, D=2048] f32  (1 GiB)
//   query:     [D] f32
// Bandwidth-bound: 2 passes over 1 GiB => ~92us floor @ 23.3 TB/s.
// Pass 1 uses V_WMMA_F32_16X16X4_F32 (f32-precise matrix path) for the
// query dot-products, with norms computed on the side from the same LDS tile.

typedef float v2f __attribute__((ext_vector_type(2)));
typedef float v8f __attribute__((ext_vector_type(8)));

#define ROWS_PER_BLOCK 128   // 8 waves x 16 rows
#define KTILE 64             // K-chunk staged in LDS per wave
#define LDS_STRIDE 68        // 64 + 4 pad -> conflict-free ds reads

// ---------------------------------------------------------------------------
// Kernel 1: dots[i] = A[i,:].q   and   norms[i] = ||A[i,:]||
// One wave computes 16 rows via WMMA f32 16x16x4 (query replicated across the
// 16 B-columns, so every column of D holds the same dot vector).
// ---------------------------------------------------------------------------
__global__ __launch_bounds__(256)
void dsdm_dots_norms(const float* __restrict__ A, const float* __restrict__ Q,
                     float* __restrict__ dots, float* __restrict__ norms,
                     int N, int D) {
    __shared__ float q_lds[2048];
    __shared__ float tile[8][16 * LDS_STRIDE];   // 8 waves, private 16xKTILE tiles

    const int lane = threadIdx.x & 31;
    const int wave = threadIdx.x >> 5;
    const int rowbase = blockIdx.x * ROWS_PER_BLOCK + wave * 16;

    // Cooperative stage of the whole query into LDS (read once per block).
    for (int i = threadIdx.x; i < D; i += blockDim.x) q_lds[i] = Q[i];
    __syncthreads();

    float* __restrict__ tp = tile[wave];

    // A fragment mapping (ISA 7.12.2, 32-bit A 16x4):
    //   lanes 0-15:  M=lane,    K = k, k+1
    //   lanes 16-31: M=lane-16, K = k+2, k+3
    const int mrow = lane & 15;
    const int koff = (lane >> 4) << 1;

    v8f c = {};          // 16x16 f32 accumulator (all columns identical)
    float sq = 0.0f;     // partial sum-of-squares for row `mrow`

    for (int k0 = 0; k0 < D; k0 += KTILE) {
        // ---- stage 16 x KTILE f32 tile: two rows per step, float4/lane ----
#pragma unroll
        for (int s = 0; s < 8; ++s) {
            const int row = 2 * s + (lane >> 4);
            const int col = (lane & 15) << 2;
            const size_t gidx = (size_t)(rowbase + row) * (size_t)D + (size_t)(k0 + col);
            const float4 g = *(const float4*)(A + gidx);
            *(float4*)&tp[row * LDS_STRIDE + col] = g;       // ds_store_b128
            if (k0 + KTILE < D)                               // warm next chunk
                __builtin_prefetch(A + gidx + KTILE, 0, 3);   // global_prefetch_b8
        }
        // No __syncthreads: each wave consumes only its own tile.

        // ---- consume: KTILE/4 WMMAs, norms piggyback on the A reads ----
#pragma unroll 4
        for (int kk = 0; kk < KTILE; kk += 4) {
            const int ao = mrow * LDS_STRIDE + kk + koff;     // 8B aligned
            v2f a; a.x = tp[ao];             a.y = tp[ao + 1];          // ds_load_b64
            v2f b; b.x = q_lds[k0 + kk + koff]; b.y = q_lds[k0 + kk + koff + 1];
            sq = fmaf(a.x, a.x, sq);
            sq = fmaf(a.y, a.y, sq);
            // D = A(16x4) x B(4x16) + C ; f32 in, f32 out.
            c = __builtin_amdgcn_wmma_f32_16x16x4_f32(
                    /*neg_a=*/false, a, /*neg_b=*/false, b,
                    /*c_mod=*/(short)0, c, /*reuse_a=*/false, /*reuse_b=*/false);
        }
    }

    // lane l and lane l+16 together hold all K of row (l&15)
    float tot = sq + __shfl_xor(sq, 16, 32);
    if (lane < 16) norms[rowbase + lane] = sqrtf(tot);

    // D-matrix layout: VGPR v, lanes 0-15 -> M=v, lanes 16-31 -> M=8+v.
    if (lane == 0) {
#pragma unroll
        for (int v = 0; v < 8; ++v) dots[rowbase + v] = c[v];
    } else if (lane == 16) {
#pragma unroll
        for (int v = 0; v < 8; ++v) dots[rowbase + 8 + v] = c[v];
    }
}

// ---------------------------------------------------------------------------
// Kernel 2: single block. qnorm, sims = dots / max(norm*qnorm, eps),
// stable softmax of (sim-1)/T  (== softmin of distances). Also zeroes d_out.
// vec[] is overwritten in place: dots -> sims -> unnormalized weights.
// Writes sum of weights to scalars[0].
// ---------------------------------------------------------------------------
__global__ __launch_bounds__(1024)
void dsdm_softmax(const float* __restrict__ Q,
                  float* __restrict__ vec,
                  const float* __restrict__ norms,
                  float* __restrict__ scalars,
                  float* __restrict__ out,
                  int N, int D) {
    __shared__ float sh[1024];
    __shared__ float s_qnorm, s_max;
    const int t = threadIdx.x;
    const int nt = blockDim.x;

    // zero the output accumulator for the atomic pass
    for (int i = t; i < D; i += nt) out[i] = 0.0f;

    // ||q||
    float qs = 0.0f;
    for (int i = t; i < D; i += nt) { const float v = Q[i]; qs = fmaf(v, v, qs); }
    sh[t] = qs; __syncthreads();
    for (int s = nt >> 1; s > 0; s >>= 1) { if (t < s) sh[t] += sh[t + s]; __syncthreads(); }
    if (t == 0) s_qnorm = sqrtf(sh[0]);
    __syncthreads();
    const float qn = s_qnorm;

    // sims + running max
    float mx = -3.402823466e38f;
    for (int i = t; i < N; i += nt) {
        const float sim = vec[i] / fmaxf(norms[i] * qn, 1e-8f);
        vec[i] = sim;
        mx = fmaxf(mx, sim);
    }
    sh[t] = mx; __syncthreads();
    for (int s = nt >> 1; s > 0; s >>= 1) { if (t < s) sh[t] = fmaxf(sh[t], sh[t + s]); __syncthreads(); }
    if (t == 0) s_max = sh[0];
    __syncthreads();
    const float smax = s_max;

    // exp((sim - smax)/T), T = 0.1
    const float invT = 10.0f;
    float tot = 0.0f;
    for (int i = t; i < N; i += nt) {
        const float w = expf((vec[i] - smax) * invT);
        vec[i] = w;
        tot += w;
    }
    sh[t] = tot; __syncthreads();
    for (int s = nt >> 1; s > 0; s >>= 1) { if (t < s) sh[t] += sh[t + s]; __syncthreads(); }
    if (t == 0) scalars[0] = sh[0];
}

// ---------------------------------------------------------------------------
// Kernel 3: retrieved[d] = (1/sum_w) * sum_i w[i] * A[i,d]
// 256 threads/block, 128 rows/block, thread t owns cols {4t..4t+3, 1024+4t..}.
// Streaming float4 loads, register accumulation, one atomicAdd burst at end.
// ---------------------------------------------------------------------------
__global__ __launch_bounds__(256)
void dsdm_retrieve(const float* __restrict__ A,
                   const float* __restrict__ w,
                   const float* __restrict__ scalars,
                   float* __restrict__ out,
                   int N, int D, int rows_per_block) {
    const int t = threadIdx.x;
    const int r0 = blockIdx.x * rows_per_block;
    const float inv = 1.0f / scalars[0];

    float4 acc0 = make_float4(0.f, 0.f, 0.f, 0.f);
    float4 acc1 = make_float4(0.f, 0.f, 0.f, 0.f);

    for (int r = 0; r < rows_per_block; ++r) {
        const float wt = w[r0 + r];
        const float4* row = (const float4*)(A + (size_t)(r0 + r) * (size_t)D);
        const float4 a0 = row[t];          // cols 4t .. 4t+3
        const float4 a1 = row[t + 256];    // cols 1024+4t ..
        acc0.x = fmaf(wt, a0.x, acc0.x); acc0.y = fmaf(wt, a0.y, acc0.y);
        acc0.z = fmaf(wt, a0.z, acc0.z); acc0.w = fmaf(wt, a0.w, acc0.w);
        acc1.x = fmaf(wt, a1.x, acc1.x); acc1.y = fmaf(wt, a1.y, acc1.y);
        acc1.z = fmaf(wt, a1.z, acc1.z); acc1.w = fmaf(wt, a1.w, acc1.w);
    }

    const int c0 = t << 2;
    const int c1 = (t + 256) << 2;
    atomicAdd(&out[c0 + 0], acc0.x * inv);
    atomicAdd(&out[c0 + 1], acc0.y * inv);
    atomicAdd(&out[c0 + 2], acc0.z * inv);
    atomicAdd(&out[c0 + 3], acc0.w * inv);
    atomicAdd(&out[c1 + 0], acc1.x * inv);
    atomicAdd(&out[c1 + 1], acc1.y * inv);
    atomicAdd(&out[c1 + 2], acc1.z * inv);
    atomicAdd(&out[c1 + 3], acc1.w * inv);
}

// ---------------------------------------------------------------------------
extern "C" void kernel_launch(void* const* d_in, const int* in_sizes, int n_in,
                              void* d_out, int out_size, void* d_ws, size_t ws_size,
                              hipStream_t stream) {
    const float* A = (const float*)d_in[0];   // [N, D] f32
    const float* Q = (const float*)d_in[1];   // [D] f32
    const int D = in_sizes[1];                // 2048
    const int N = in_sizes[0] / D;            // 131072
    float* out = (float*)d_out;               // [D] f32

    // workspace: dots/sims/weights (N), norms (N), scalars
    float* dots    = (float*)d_ws;
    float* norms   = dots + N;
    float* scalars = norms + N;

    dsdm_dots_norms<<<dim3(N / ROWS_PER_BLOCK), dim3(256), 0, stream>>>(
        A, Q, dots, norms, N, D);

    dsdm_softmax<<<dim3(1), dim3(1024), 0, stream>>>(
        Q, dots, norms, scalars, out, N, D);

    dsdm_retrieve<<<dim3(N / ROWS_PER_BLOCK), dim3(256), 0, stream>>>(
        A, dots, scalars, out, N, D, ROWS_PER_BLOCK);
}